// WaveNetBackbone_13580686590252
// MI455X (gfx1250) — compile-verified
//
#include <hip/hip_runtime.h>

// ---------------------------------------------------------------------------
// WaveNet backbone on MI455X (gfx1250), fp32 WMMA (V_WMMA_F32_16X16X4_F32).
// B=2, C=S=128, T=8192, L=20 layers, dilations 1..512 repeated twice.
// One fused kernel per layer. Each wave computes a 16(m) x 32(n) output strip
// with two accumulators sharing each A (weight) fragment -> halves A traffic
// per WMMA. Intermediates (x_dil, x_h) live only in LDS/registers.
// ---------------------------------------------------------------------------

typedef float v2f __attribute__((ext_vector_type(2)));
typedef float v8f __attribute__((ext_vector_type(8)));

#define TT      8192
#define CCH     128
#define NB      2
#define NLAYERS 20
#define NT      32            // time columns per workgroup
#define LDB     33            // LDS row stride (pad to dodge bank conflicts)
#define BCT     (NB * CCH * TT)   // 2,097,152 elements per [B,C,T] tensor

// ---- dual-tile GEMM: D{0,1} += A(16xK) * B{0,1}(Kx16), K = 4*KS ------------
// Arow : &A[(m0 + lane%16) * lda]  (global, row-major, k contiguous)
// Bc0/1: &Blds[n0{a,b} + lane%16]  (LDS, row stride LDB)
// kb   : 2*(lane>>4)  per-lane k sub-offset (ISA A/B fragment layout)
template <int KS>
__device__ __forceinline__ void gemm2(const float* __restrict__ Arow,
                                      const float* Bc0, const float* Bc1,
                                      int kb, v8f& acc0, v8f& acc1) {
#pragma unroll 4
  for (int ks = 0; ks < KS; ++ks) {
    const int k = ks * 4 + kb;
    v2f a = *(const v2f*)(Arow + k);         // global_load_b64, shared by 2 WMMA
    v2f b0; b0.x = Bc0[k * LDB]; b0.y = Bc0[(k + 1) * LDB];
    v2f b1; b1.x = Bc1[k * LDB]; b1.y = Bc1[(k + 1) * LDB];
    acc0 = __builtin_amdgcn_wmma_f32_16x16x4_f32(false, a, false, b0,
                                                 (short)0, acc0, false, false);
    acc1 = __builtin_amdgcn_wmma_f32_16x16x4_f32(false, a, false, b1,
                                                 (short)0, acc1, false, false);
  }
}

// ---- fused per-layer kernel ------------------------------------------------
__global__ __launch_bounds__(256)
void wavenet_layer(const float* __restrict__ h_in,   // [B][C][T]
                   float* __restrict__ h_out,        // [B][C][T] (ping-pong)
                   float* __restrict__ out_next,     // outputs[i+1] slot or null
                   float* __restrict__ skip_out,     // skips[i] slot [B][S][T]
                   const float* __restrict__ Adil,   // [128][256] pre-transposed
                   const float* __restrict__ bdil,
                   const float* __restrict__ Wt, const float* __restrict__ bt,
                   const float* __restrict__ Wsg, const float* __restrict__ bsg,
                   const float* __restrict__ Wk, const float* __restrict__ bk,
                   int dil) {
  __shared__ float smem[256 * LDB];        // 33,792 B
  float* bin  = smem;                      // phase 1: 256 x LDB input tile
  float* xdil = smem;                      // phase 2: 128 x LDB (aliases rows 0-127)
  float* xh   = smem + CCH * LDB;          // phase 2/3: 128 x LDB (aliases rows 128-255)

  const int tid  = threadIdx.x;
  const int lane = tid & 31;
  const int wv   = tid >> 5;               // 0..7
  const int m0   = wv * 16;                // output-channel tile (whole M covered)
  const int lmod = lane & 15;
  const int kb   = (lane >> 4) * 2;        // per-lane k sub-offset
  const int mhi  = (lane >> 4) * 8;        // C/D fragment row offset

  const int b  = blockIdx.y;
  const int t0 = blockIdx.x * NT;

  const float* Bc0 = nullptr;              // set per phase
  const float* Bc1 = nullptr;

  // -- stage input: rows 0..127 = h[c, t-d] (causal zero pad), 128..255 = h[c, t]
  const float* hb = h_in + (size_t)b * (CCH * TT);
  for (int idx = tid; idx < 256 * NT; idx += 256) {
    const int row = idx >> 5, col = idx & 31;
    float v;
    if (row < CCH) {
      const int tc = t0 + col - dil;
      v = (tc >= 0) ? hb[row * TT + tc] : 0.0f;
    } else {
      v = hb[(row - CCH) * TT + (t0 + col)];
    }
    bin[row * LDB + col] = v;
  }
  __syncthreads();

  // -- phase 1: x_dil = Adil(128x256) @ [h_shift; h] + bdil  (2 x 64 wmma)
  v8f d0 = {}, d1 = {};
  Bc0 = bin + lmod; Bc1 = bin + 16 + lmod;
  gemm2<64>(Adil + (m0 + lmod) * 256, Bc0, Bc1, kb, d0, d1);
#pragma unroll
  for (int r = 0; r < 8; ++r) {
    const float bv = bdil[m0 + r + mhi];
    d0[r] += bv; d1[r] += bv;
  }
  __syncthreads();                         // all reads of bin complete
#pragma unroll
  for (int r = 0; r < 8; ++r) {
    const int row = (m0 + r + mhi) * LDB;
    xdil[row + lmod]      = d0[r];
    xdil[row + 16 + lmod] = d1[r];
  }
  __syncthreads();

  // -- phase 2: filt = tanh(Wt @ x_dil + bt), gate = sig(Ws @ x_dil + bs)
  v8f t0a = {}, t1a = {}, s0a = {}, s1a = {};
  Bc0 = xdil + lmod; Bc1 = xdil + 16 + lmod;
  gemm2<32>(Wt  + (m0 + lmod) * CCH, Bc0, Bc1, kb, t0a, t1a);
  gemm2<32>(Wsg + (m0 + lmod) * CCH, Bc0, Bc1, kb, s0a, s1a);
  v8f xh0, xh1;
#pragma unroll
  for (int r = 0; r < 8; ++r) {
    const float btv = bt[m0 + r + mhi];
    const float bsv = bsg[m0 + r + mhi];
    const float f0 = tanhf(t0a[r] + btv);
    const float f1 = tanhf(t1a[r] + btv);
    const float g0 = 1.0f / (1.0f + __expf(-(s0a[r] + bsv)));
    const float g1 = 1.0f / (1.0f + __expf(-(s1a[r] + bsv)));
    xh0[r] = f0 * g0;
    xh1[r] = f1 * g1;
  }
#pragma unroll
  for (int r = 0; r < 8; ++r) {            // xh region disjoint from xdil
    const int row = (m0 + r + mhi) * LDB;
    xh[row + lmod]      = xh0[r];
    xh[row + 16 + lmod] = xh1[r];
  }
  __syncthreads();

  // -- phase 3: skip = Wk @ x_h + bk; h_next = x_h + x_dil
  v8f k0 = {}, k1 = {};
  Bc0 = xh + lmod; Bc1 = xh + 16 + lmod;
  gemm2<32>(Wk + (m0 + lmod) * CCH, Bc0, Bc1, kb, k0, k1);

  const int tc0 = t0 + lmod;
  const int tc1 = t0 + 16 + lmod;
  float* skb = skip_out + (size_t)b * (CCH * TT);
  float* hob = h_out + (size_t)b * (CCH * TT);
#pragma unroll
  for (int r = 0; r < 8; ++r) {
    const int o = m0 + r + mhi;
    const float bkv = bk[o];
    skb[o * TT + tc0] = k0[r] + bkv;
    skb[o * TT + tc1] = k1[r] + bkv;
    hob[o * TT + tc0] = xh0[r] + d0[r];
    hob[o * TT + tc1] = xh1[r] + d1[r];
  }
  if (out_next) {                          // uniform branch
    float* onb = out_next + (size_t)b * (CCH * TT);
#pragma unroll
    for (int r = 0; r < 8; ++r) {
      const int o = m0 + r + mhi;
      onb[o * TT + tc0] = xh0[r] + d0[r];
      onb[o * TT + tc1] = xh1[r] + d1[r];
    }
  }
}

// ---- input projection: h = w_in[c]*x[b,t] + b_in[c]; also writes outputs[0]
__global__ void input_proj(const float* __restrict__ x,
                           const float* __restrict__ w_in,
                           const float* __restrict__ b_in,
                           float* __restrict__ h, float* __restrict__ out0) {
  const int i = blockIdx.x * blockDim.x + threadIdx.x;   // over B*C*T
  if (i >= BCT) return;
  const int t  = i % TT;
  const int c  = (i / TT) % CCH;
  const int bb = i / (CCH * TT);
  const float v = w_in[c] * x[bb * TT + t] + b_in[c];
  h[i]    = v;
  out0[i] = v;
}

// ---- build per-layer [128 x 256] dilated weight: cols 0..127 = tap0, 128..255 = tap1
__global__ void prep_adil(const float* __restrict__ w_dil, float* __restrict__ Adil) {
  const int i = blockIdx.x * blockDim.x + threadIdx.x;   // over L*128*256
  if (i >= NLAYERS * CCH * 256) return;
  const int k = i % 256;
  const int o = (i / 256) % CCH;
  const int l = i / (256 * CCH);
  const int c   = (k < CCH) ? k : (k - CCH);
  const int tap = (k < CCH) ? 0 : 1;
  Adil[i] = w_dil[((l * CCH + o) * CCH + c) * 2 + tap];
}

// ---------------------------------------------------------------------------
extern "C" void kernel_launch(void* const* d_in, const int* in_sizes, int n_in,
                              void* d_out, int out_size, void* d_ws, size_t ws_size,
                              hipStream_t stream) {
  const float* x     = (const float*)d_in[0];
  const float* w_in  = (const float*)d_in[1];
  const float* b_in  = (const float*)d_in[2];
  const float* w_dil = (const float*)d_in[3];
  const float* b_dil = (const float*)d_in[4];
  const float* w_t   = (const float*)d_in[5];
  const float* b_t   = (const float*)d_in[6];
  const float* w_s   = (const float*)d_in[7];
  const float* b_s   = (const float*)d_in[8];
  const float* w_k   = (const float*)d_in[9];
  const float* b_k   = (const float*)d_in[10];
  float* out = (float*)d_out;

  float* ws   = (float*)d_ws;
  float* h0   = ws;                      // [B][C][T]
  float* h1   = ws + BCT;                // [B][C][T]
  float* adil = ws + 2 * (size_t)BCT;    // [L][128][256]

  {
    const int n = NLAYERS * CCH * 256;
    prep_adil<<<(n + 255) / 256, 256, 0, stream>>>(w_dil, adil);
  }
  {
    const int n = BCT;
    input_proj<<<(n + 255) / 256, 256, 0, stream>>>(x, w_in, b_in, h0,
                                                    out /* outputs[0] */);
  }

  static const int DIL[NLAYERS] = {1, 2, 4, 8, 16, 32, 64, 128, 256, 512,
                                   1, 2, 4, 8, 16, 32, 64, 128, 256, 512};

  float* hin  = h0;
  float* hout = h1;
  for (int i = 0; i < NLAYERS; ++i) {
    float* out_next = (i + 1 < NLAYERS) ? (out + (size_t)(i + 1) * BCT) : nullptr;
    float* skip     = out + (size_t)NLAYERS * BCT + (size_t)i * BCT;
    wavenet_layer<<<dim3(TT / NT, NB), 256, 0, stream>>>(
        hin, hout, out_next, skip,
        adil + (size_t)i * CCH * 256, b_dil + i * CCH,
        w_t + (size_t)i * CCH * CCH, b_t + i * CCH,
        w_s + (size_t)i * CCH * CCH, b_s + i * CCH,
        w_k + (size_t)i * CCH * CCH, b_k + i * CCH,
        DIL[i]);
    float* tmp = hin; hin = hout; hout = tmp;
  }
}